// Net_86595130622536
// MI455X (gfx1250) — compile-verified
//
#include <hip/hip_runtime.h>
#include <hip/hip_bf16.h>
#include <math.h>

#define NNODES 100000
#define NFEAT  512
#define NHID   16
#define NCLASS 40
#define NEDGE  3200000

typedef __attribute__((ext_vector_type(2))) float v2f;
typedef __attribute__((ext_vector_type(8))) float v8f;

// ---------------------------------------------------------------------------
// Monotone float atomic-max via int/uint ordering trick (always compiles,
// lowers to global_atomic_max_i32 / global_atomic_min_u32).
// ---------------------------------------------------------------------------
__device__ __forceinline__ void atomicMaxF32(float* addr, float v) {
    if (v >= 0.0f) {
        atomicMax((int*)addr, __float_as_int(v));
    } else {
        atomicMin((unsigned int*)addr, (unsigned int)__float_as_int(v));
    }
}

// ---------------------------------------------------------------------------
// Kernel 1: h = relu(x @ W1^T + b1)  via V_WMMA_F32_16X16X4_F32
//   - one wave per 16-row output tile (8 waves / block -> 128 rows / block)
//   - W1 staged transposed into LDS as Wt[k][n] (conflict-free B reads)
//   - bias folded into accumulator init; uniform fast-path epilogue
// ---------------------------------------------------------------------------
__launch_bounds__(256)
__global__ void gemm1_relu_wmma(const float* __restrict__ x,
                                const float* __restrict__ w1,
                                const float* __restrict__ b1,
                                float* __restrict__ h) {
    __shared__ float Wt[NFEAT * NHID];  // [k][n] transposed, 32 KB

    for (int t = threadIdx.x; t < NFEAT * NHID; t += 256) {
        int n = t >> 9;      // w1 is [16][512] row-major
        int k = t & 511;
        Wt[k * NHID + n] = w1[t];
    }
    __syncthreads();

    const int wave  = threadIdx.x >> 5;   // wave32
    const int lane  = threadIdx.x & 31;
    const int n     = lane & 15;          // N index (and M index for A loads)
    const int khalf = lane >> 4;          // 0: K=0,1   1: K=2,3

    const int rowBase = (blockIdx.x * 8 + wave) * 16;
    int row  = rowBase + n;
    int rowc = row < NNODES ? row : (NNODES - 1);  // clamp loads, guard stores
    const float* xrow = x + (size_t)rowc * NFEAT;

    const float bias = b1[n];
    v8f c;
    #pragma unroll
    for (int v = 0; v < 8; ++v) c[v] = bias;       // D[m][n] starts at b1[n]

    #pragma unroll 4
    for (int kk = 0; kk < NFEAT; kk += 4) {
        const int k0 = kk + khalf * 2;
        float2 av = *(const float2*)(xrow + k0);   // global_load_b64
        v2f a, b;
        a.x = av.x; a.y = av.y;
        b.x = Wt[k0 * NHID + n];                   // ds_load_2addr, no conflicts
        b.y = Wt[(k0 + 1) * NHID + n];
        c = __builtin_amdgcn_wmma_f32_16x16x4_f32(
                false, a, false, b, (short)0, c, false, false);
    }

    if (rowBase + 16 <= NNODES) {                  // uniform fast path
        #pragma unroll
        for (int v = 0; v < 8; ++v) {
            int m = v + 8 * khalf;                 // C/D layout: M = vgpr + 8*half
            float val = c[v];
            h[(size_t)(rowBase + m) * NHID + n] = val > 0.0f ? val : 0.0f;
        }
    } else {                                       // only the last block
        #pragma unroll
        for (int v = 0; v < 8; ++v) {
            int m = v + 8 * khalf;
            int r = rowBase + m;
            if (r < NNODES) {
                float val = c[v];
                h[(size_t)r * NHID + n] = val > 0.0f ? val : 0.0f;
            }
        }
    }
}

// ---------------------------------------------------------------------------
// Kernel 2: row-wise L2 normalization, xn[i] = h[i] / max(||h[i]||, 1e-12)
// ---------------------------------------------------------------------------
__launch_bounds__(256)
__global__ void normalize_kernel(const float* __restrict__ feat,
                                 float* __restrict__ xn) {
    int i = blockIdx.x * 256 + threadIdx.x;
    if (i >= NNODES) return;
    const float4* p = (const float4*)(feat + (size_t)i * NHID);
    float4 a0 = p[0], a1 = p[1], a2 = p[2], a3 = p[3];
    float ss = a0.x*a0.x + a0.y*a0.y + a0.z*a0.z + a0.w*a0.w
             + a1.x*a1.x + a1.y*a1.y + a1.z*a1.z + a1.w*a1.w
             + a2.x*a2.x + a2.y*a2.y + a2.z*a2.z + a2.w*a2.w
             + a3.x*a3.x + a3.y*a3.y + a3.z*a3.z + a3.w*a3.w;
    float inv = 1.0f / fmaxf(sqrtf(ss), 1e-12f);
    float4* q = (float4*)(xn + (size_t)i * NHID);
    a0.x*=inv; a0.y*=inv; a0.z*=inv; a0.w*=inv;
    a1.x*=inv; a1.y*=inv; a1.z*=inv; a1.w*=inv;
    a2.x*=inv; a2.y*=inv; a2.z*=inv; a2.w*=inv;
    a3.x*=inv; a3.y*=inv; a3.z*=inv; a3.w*=inv;
    q[0]=a0; q[1]=a1; q[2]=a2; q[3]=a3;
}

// ---------------------------------------------------------------------------
// Kernel 3: per-layer init: m=-big, s=0, agg=0   (run every call -> stateless)
// ---------------------------------------------------------------------------
__launch_bounds__(256)
__global__ void seg_init_kernel(float* __restrict__ mbuf,
                                float* __restrict__ sbuf,
                                float* __restrict__ agg) {
    int i = blockIdx.x * 256 + threadIdx.x;
    if (i >= NNODES) return;
    mbuf[i] = -3.0e38f;
    sbuf[i] = 0.0f;
    float4 z = make_float4(0.f, 0.f, 0.f, 0.f);
    float4* p = (float4*)(agg + (size_t)i * NHID);
    p[0] = z; p[1] = z; p[2] = z; p[3] = z;
}

// ---------------------------------------------------------------------------
// Kernel 4: edge pass 1 (16 lanes per edge):
//   alpha[e] = beta * <xn[dst], xn[src]>  via coalesced 64B gathers +
//   4-step shuffle butterfly; lane 0 of each group does the segment max.
//   NEDGE % 16 == 0 and 256 % 16 == 0  ->  no tail divergence.
// ---------------------------------------------------------------------------
__launch_bounds__(256)
__global__ void edge_alpha_kernel(const int* __restrict__ src,
                                  const int* __restrict__ dst,
                                  const float* __restrict__ xn,
                                  const float* __restrict__ betaPtr,
                                  float* __restrict__ alpha,
                                  float* __restrict__ mbuf) {
    int tid = blockIdx.x * 256 + threadIdx.x;
    int e   = tid >> 4;            // one edge per 16 lanes
    int j   = threadIdx.x & 15;    // feature index owned by this lane
    if (e >= NEDGE) return;
    int s = src[e], d = dst[e];    // uniform within group -> line broadcast
    float prod = xn[(size_t)s * NHID + j] * xn[(size_t)d * NHID + j];
    // reduce across the 16-lane group (stays inside the half-wave)
    prod += __shfl_xor(prod, 1, 32);
    prod += __shfl_xor(prod, 2, 32);
    prod += __shfl_xor(prod, 4, 32);
    prod += __shfl_xor(prod, 8, 32);
    if (j == 0) {
        float beta = betaPtr ? betaPtr[0] : 1.0f;
        float a = beta * prod;
        alpha[e] = a;
        atomicMaxF32(mbuf + d, a);
    }
}

// ---------------------------------------------------------------------------
// Kernel 5: edge pass 2 (thread per edge): ev = exp(alpha - m[dst]);
//   segment sum (alpha := ev)
// ---------------------------------------------------------------------------
__launch_bounds__(256)
__global__ void edge_exp_kernel(const int* __restrict__ dst,
                                float* __restrict__ alpha,
                                const float* __restrict__ mbuf,
                                float* __restrict__ sbuf) {
    int e = blockIdx.x * 256 + threadIdx.x;
    if (e >= NEDGE) return;
    int d = dst[e];
    float ev = expf(alpha[e] - mbuf[d]);
    alpha[e] = ev;
    atomicAdd(sbuf + d, ev);
}

// ---------------------------------------------------------------------------
// Kernel 6: edge pass 3 (16 lanes per edge):
//   w = ev / max(s[dst],1e-16); agg[dst][j] += feat[src][j] * w
//   One coalesced 64B gather + one coalesced atomic per edge group.
// ---------------------------------------------------------------------------
__launch_bounds__(256)
__global__ void edge_agg_kernel(const int* __restrict__ src,
                                const int* __restrict__ dst,
                                const float* __restrict__ feat,
                                const float* __restrict__ alpha,
                                const float* __restrict__ sbuf,
                                float* __restrict__ agg) {
    int tid = blockIdx.x * 256 + threadIdx.x;
    int e   = tid >> 4;
    int j   = threadIdx.x & 15;
    if (e >= NEDGE) return;
    int s = src[e], d = dst[e];                    // broadcast within group
    float w = alpha[e] / fmaxf(sbuf[d], 1e-16f);   // broadcast within group
    float val = feat[(size_t)s * NHID + j] * w;    // coalesced 64B gather
    atomicAdd(agg + (size_t)d * NHID + j, val);    // coalesced 64B atomic
}

// ---------------------------------------------------------------------------
// Kernel 7: out = log_softmax(h @ W2^T + b2); W2/b2 staged in LDS
// ---------------------------------------------------------------------------
__launch_bounds__(256)
__global__ void head_kernel(const float* __restrict__ feat,
                            const float* __restrict__ w2,
                            const float* __restrict__ b2,
                            float* __restrict__ out) {
    __shared__ float W2s[NCLASS * NHID];
    __shared__ float B2s[NCLASS];
    for (int t = threadIdx.x; t < NCLASS * NHID; t += 256) W2s[t] = w2[t];
    if (threadIdx.x < NCLASS) B2s[threadIdx.x] = b2[threadIdx.x];
    __syncthreads();

    int i = blockIdx.x * 256 + threadIdx.x;
    if (i >= NNODES) return;

    float hv[NHID];
    const float4* p = (const float4*)(feat + (size_t)i * NHID);
    float4 a0 = p[0], a1 = p[1], a2 = p[2], a3 = p[3];
    hv[0]=a0.x; hv[1]=a0.y; hv[2]=a0.z; hv[3]=a0.w;
    hv[4]=a1.x; hv[5]=a1.y; hv[6]=a1.z; hv[7]=a1.w;
    hv[8]=a2.x; hv[9]=a2.y; hv[10]=a2.z; hv[11]=a2.w;
    hv[12]=a3.x; hv[13]=a3.y; hv[14]=a3.z; hv[15]=a3.w;

    float logits[NCLASS];
    float mx = -3.0e38f;
    #pragma unroll
    for (int c = 0; c < NCLASS; ++c) {
        float acc = B2s[c];
        #pragma unroll
        for (int k = 0; k < NHID; ++k) acc += hv[k] * W2s[c * NHID + k];
        logits[c] = acc;
        mx = fmaxf(mx, acc);
    }
    float sum = 0.0f;
    #pragma unroll
    for (int c = 0; c < NCLASS; ++c) sum += expf(logits[c] - mx);
    float lse = logf(sum);
    float* o = out + (size_t)i * NCLASS;
    #pragma unroll
    for (int c = 0; c < NCLASS; ++c) o[c] = logits[c] - mx - lse;
}

// ---------------------------------------------------------------------------
// Host-side launcher
// ---------------------------------------------------------------------------
extern "C" void kernel_launch(void* const* d_in, const int* in_sizes, int n_in,
                              void* d_out, int out_size, void* d_ws, size_t ws_size,
                              hipStream_t stream) {
    const float* x     = (const float*)d_in[0];            // [N, 512]
    const int*   eidx  = (const int*)d_in[1];              // [2, E]
    const float* w1    = (const float*)d_in[2];            // [16, 512]
    const float* b1    = (const float*)d_in[3];            // [16]
    const float* beta2 = (const float*)d_in[4];            // [1]
    const float* w2    = (const float*)d_in[5];            // [40, 16]
    const float* b2    = (const float*)d_in[6];            // [40]
    float*       out   = (float*)d_out;                    // [N, 40]

    const int* src = eidx;
    const int* dst = eidx + NEDGE;

    // workspace layout (floats): h | xn | agg | m | s | alpha
    float* h    = (float*)d_ws;
    float* xn   = h   + (size_t)NNODES * NHID;
    float* agg  = xn  + (size_t)NNODES * NHID;
    float* mbuf = agg + (size_t)NNODES * NHID;
    float* sbuf = mbuf + NNODES;
    float* abuf = sbuf + NNODES;                           // [E]

    const int NB_NODE  = (NNODES + 255) / 256;             // 391
    const int NB_EDGE  = (NEDGE + 255) / 256;              // 12500 (1 thread/edge)
    const int NB_EDGE16 = NEDGE / 16;                      // 200000 (16 lanes/edge)
    const int NB_GEMM  = (NNODES + 127) / 128;             // 782 (8 waves x 16 rows)

    // h = relu(x @ W1^T + b1)
    gemm1_relu_wmma<<<NB_GEMM, 256, 0, stream>>>(x, w1, b1, h);

    // ---- AGNN layer 1 (beta = 1), h -> agg ----
    normalize_kernel<<<NB_NODE, 256, 0, stream>>>(h, xn);
    seg_init_kernel<<<NB_NODE, 256, 0, stream>>>(mbuf, sbuf, agg);
    edge_alpha_kernel<<<NB_EDGE16, 256, 0, stream>>>(src, dst, xn, nullptr, abuf, mbuf);
    edge_exp_kernel<<<NB_EDGE, 256, 0, stream>>>(dst, abuf, mbuf, sbuf);
    edge_agg_kernel<<<NB_EDGE16, 256, 0, stream>>>(src, dst, h, abuf, sbuf, agg);

    // ---- AGNN layer 2 (beta = beta2[0]), agg -> h ----
    normalize_kernel<<<NB_NODE, 256, 0, stream>>>(agg, xn);
    seg_init_kernel<<<NB_NODE, 256, 0, stream>>>(mbuf, sbuf, h);
    edge_alpha_kernel<<<NB_EDGE16, 256, 0, stream>>>(src, dst, xn, beta2, abuf, mbuf);
    edge_exp_kernel<<<NB_EDGE, 256, 0, stream>>>(dst, abuf, mbuf, sbuf);
    edge_agg_kernel<<<NB_EDGE16, 256, 0, stream>>>(src, dst, agg, abuf, sbuf, h);

    // logits + log_softmax
    head_kernel<<<NB_NODE, 256, 0, stream>>>(h, w2, b2, out);
}